// TopicFormer_4303557230899
// MI455X (gfx1250) — compile-verified
//
#include <hip/hip_runtime.h>

// Problem constants (reference: N, L, C, K = 8, 6400, 256, 100)
#define N_   8
#define L_   6400
#define C_   256
#define K_   100
#define KPAD 112                      // 7 WMMA tiles of 16 columns
#define SCH  32                       // L split into 32 chunks for parallelism
#define ROWS_PER_CHUNK (L_ / SCH)     // 200 rows, multiple of 4

typedef __attribute__((ext_vector_type(2))) float v2f;
typedef __attribute__((ext_vector_type(8))) float v8f;

// ---------------------------------------------------------------------------
// K1: partial column sums of prob_topics halves via ones-GEMV on the WMMA pipe.
// D = A(16x4, all ones) x B(4x16 chunk of P) + C  accumulates 4 L-rows across
// 16 K-columns per v_wmma_f32_16x16x4_f32. Result broadcast across M; lanes
// 0..15, acc VGPR0 hold the 16 column sums.
// grid = (SCH, N, 2), block = 128 (4 waves; wave w owns tiles {w, w+4}).
// ---------------------------------------------------------------------------
__global__ __launch_bounds__(128) void k1_partial_sums(
    const float* __restrict__ prob, float* __restrict__ part)
{
    const int s = blockIdx.x, n = blockIdx.y, h = blockIdx.z;
    const int lane = threadIdx.x & 31;
    const int wave = threadIdx.x >> 5;
    const int ncol = lane & 15;
    const int half = lane >> 4;

    const float* __restrict__ P =
        prob + ((size_t)n * 2 * L_ + (size_t)h * L_) * K_;
    float* __restrict__ pb =
        part + (((size_t)h * N_ + n) * SCH + s) * KPAD;

    v2f a; a.x = 1.0f; a.y = 1.0f;   // ones A-matrix (all 64 entries == 1.0)

    for (int ti = 0; ti < 2; ++ti) {
        const int t = wave + ti * 4;            // tile id, uniform per wave
        if (t >= 7) break;
        const int keff  = t * 16 + ncol;        // logical K column
        const int kc    = keff < K_ ? keff : (K_ - 1);  // clamped (no div.)
        const float sel = keff < K_ ? 1.0f : 0.0f;      // zero pad columns

        v8f acc = {};
        const int lbase = s * ROWS_PER_CHUNK;
        for (int g = 0; g < ROWS_PER_CHUNK; g += 4) {
            const int row0 = lbase + g + 2 * half;
            v2f b;
            b.x = P[(size_t)row0 * K_ + kc] * sel;
            b.y = P[(size_t)(row0 + 1) * K_ + kc] * sel;
            acc = __builtin_amdgcn_wmma_f32_16x16x4_f32(
                false, a, false, b, (short)0, acc, false, false);
        }
        if (lane < 16 && keff < K_) pb[keff] = acc[0];
    }
}

// ---------------------------------------------------------------------------
// K2: reduce the 32 partials per (h, n, k), form prod = s0*s1, argmax over k.
// Tie-break: lowest index (matches jax.lax.top_k ordering). Deterministic tree.
// grid = N, block = 128.
// ---------------------------------------------------------------------------
__global__ __launch_bounds__(128) void k2_argmax(
    const float* __restrict__ part, int* __restrict__ kstar)
{
    const int n = blockIdx.x;
    const int t = threadIdx.x;

    float val = -1.0f;                          // products are >= 0
    if (t < K_) {
        const float* p0 = part + (((size_t)0 * N_ + n) * SCH) * KPAD + t;
        const float* p1 = part + (((size_t)1 * N_ + n) * SCH) * KPAD + t;
        float s0 = 0.0f, s1 = 0.0f;
        for (int s = 0; s < SCH; ++s) {
            s0 += p0[(size_t)s * KPAD];
            s1 += p1[(size_t)s * KPAD];
        }
        val = s0 * s1;
    }

    __shared__ float sv[128];
    __shared__ int   si[128];
    sv[t] = val; si[t] = t;
    __syncthreads();
    for (int off = 64; off > 0; off >>= 1) {
        if (t < off) {
            const float vo = sv[t + off];
            const int   io = si[t + off];
            if (vo > sv[t] || (vo == sv[t] && io < si[t])) {
                sv[t] = vo; si[t] = io;
            }
        }
        __syncthreads();
    }
    if (t == 0) kstar[n] = si[0];
}

// ---------------------------------------------------------------------------
// K3: topick = topics[n, l, k*] > 0.5 ; block-sequential prefix sum over L,
// emit compaction source indices, length, and the new_mask output.
// grid = N, block = 256 (25 chunks of 256).
// ---------------------------------------------------------------------------
__global__ __launch_bounds__(256) void k3_scan(
    const float* __restrict__ topics, const int* __restrict__ kstar,
    int* __restrict__ comp_src, int* __restrict__ len_out,
    float* __restrict__ mask_out)
{
    const int n = blockIdx.x;
    const int t = threadIdx.x;
    const int ks = kstar[n];
    const float* __restrict__ T = topics + ((size_t)n * 2 * L_) * K_ + ks;

    __shared__ int sc[256];
    __shared__ int s_total;
    int running = 0;

    for (int base = 0; base < L_; base += 256) {
        const int l = base + t;
        const int flag = (T[(size_t)l * K_] > 0.5f) ? 1 : 0;
        sc[t] = flag;
        __syncthreads();
        // Hillis-Steele inclusive scan over 256 entries
        for (int off = 1; off < 256; off <<= 1) {
            const int v   = sc[t];
            const int add = (t >= off) ? sc[t - off] : 0;
            __syncthreads();
            sc[t] = v + add;
            __syncthreads();
        }
        const int incl = sc[t];
        if (t == 255) s_total = incl;
        if (flag) comp_src[(size_t)n * L_ + running + (incl - flag)] = l;
        __syncthreads();
        running += s_total;
        __syncthreads();
    }

    if (t == 0) len_out[n] = running;
    for (int base = 0; base < L_; base += 256) {
        const int l = base + t;
        mask_out[(size_t)n * L_ + l] = (l < running) ? 1.0f : 0.0f;
    }
}

// ---------------------------------------------------------------------------
// K4: compaction gather. Output row j gets feat row comp_src[j] if j < len,
// else zeros (d_out is poisoned before timing, so every row must be written).
// 64 threads x float4 = one 1 KB row per block (global_load/store_b128).
// grid = (L, N), block = 64.
// ---------------------------------------------------------------------------
__global__ __launch_bounds__(64) void k4_gather(
    const float* __restrict__ feat, const int* __restrict__ comp_src,
    const int* __restrict__ len_in, float* __restrict__ out)
{
    const int j = blockIdx.x, n = blockIdx.y, t = threadIdx.x;
    const int len = len_in[n];
    float4* dst = (float4*)(out + ((size_t)n * L_ + j) * C_);
    if (j < len) {
        const int src = comp_src[(size_t)n * L_ + j];
        const float4* s4 = (const float4*)(feat + ((size_t)n * L_ + src) * C_);
        dst[t] = s4[t];
    } else {
        dst[t] = make_float4(0.0f, 0.0f, 0.0f, 0.0f);
    }
}

// ---------------------------------------------------------------------------
extern "C" void kernel_launch(void* const* d_in, const int* in_sizes, int n_in,
                              void* d_out, int out_size, void* d_ws, size_t ws_size,
                              hipStream_t stream)
{
    (void)in_sizes; (void)n_in; (void)out_size; (void)ws_size;

    const float* feat   = (const float*)d_in[0];   // (N, L, C)
    const float* prob   = (const float*)d_in[1];   // (N, 2L, K)
    const float* topics = (const float*)d_in[2];   // (N, 2L, K)
    // d_in[3] = n_samples: only sampled_inds[:,0] affects the outputs -> unused

    float* out      = (float*)d_out;               // resized_feat | new_mask
    float* mask_out = out + (size_t)N_ * L_ * C_;

    char* ws = (char*)d_ws;
    float* part = (float*)ws;                                  // [2][N][SCH][KPAD]
    const size_t part_bytes = (size_t)2 * N_ * SCH * KPAD * sizeof(float);
    int* kstar    = (int*)(ws + part_bytes);                   // [N]
    int* len_ws   = kstar + N_;                                // [N]
    int* comp_src = len_ws + N_;                               // [N][L]

    k1_partial_sums<<<dim3(SCH, N_, 2), 128, 0, stream>>>(prob, part);
    k2_argmax      <<<N_, 128, 0, stream>>>(part, kstar);
    k3_scan        <<<N_, 256, 0, stream>>>(topics, kstar, comp_src, len_ws, mask_out);
    k4_gather      <<<dim3(L_, N_), 64, 0, stream>>>(feat, comp_src, len_ws, out);
}